// CustomLoss_14783277432960
// MI455X (gfx1250) — compile-verified
//
#include <hip/hip_runtime.h>
#include <cstdint>

// ---------------------------------------------------------------------------
// CustomLoss (cosh spatial-weight update) for MI455X / gfx1250, wave32.
//   grid = 2048 rows, block = 256 threads (8 wave32), 32 elements/thread in
//   VGPRs.  Lambda row is DMA'd into LDS by the Tensor Data Mover while the
//   20-iteration Newton solve runs on v_exp_f32/v_rcp_f32.
// ---------------------------------------------------------------------------

static constexpr int BROWS = 2048;
static constexpr int NCOLS = 8192;
static constexpr int TPB   = 256;
static constexpr int NW    = TPB / 32;          // 8 waves per block
static constexpr int CH    = NCOLS / (TPB * 4); // 8 float4 chunks per thread

typedef unsigned int u32;
typedef __attribute__((ext_vector_type(4))) u32 u32x4;
typedef __attribute__((ext_vector_type(8))) u32 u32x8;

__device__ __forceinline__ float wave_sum(float v) {
#pragma unroll
  for (int off = 16; off > 0; off >>= 1) v += __shfl_xor(v, off, 32);
  return v;
}
__device__ __forceinline__ float wave_max(float v) {
#pragma unroll
  for (int off = 16; off > 0; off >>= 1) v = fmaxf(v, __shfl_xor(v, off, 32));
  return v;
}

// u >= 0; accumulate (e^u - e^-u) and (e^u + e^-u)*u  (0.5 factored out later)
__device__ __forceinline__ void sinhcosh_acc(float rr, float inv_eps,
                                             float& s1, float& s2) {
  float u  = rr * inv_eps;
  float e  = __expf(u);
  float em = __builtin_amdgcn_rcpf(e);
  s1 += (e - em);
  s2 += (e + em) * u;
}

__device__ __forceinline__ float sinh_q(float rr, float beta) {
  float u = rr * beta;
  float e = __expf(u);
  return 0.5f * (e - __builtin_amdgcn_rcpf(e));
}

__global__ __launch_bounds__(TPB)
void cosh_weight_rows(const float* __restrict__ yp,
                      const float* __restrict__ yt,
                      const float* __restrict__ lam,
                      float* __restrict__ out_lam,   // d_out + 1
                      float* __restrict__ row_loss)  // d_ws, BROWS floats
{
  __shared__ float lam_lds[NCOLS];   // 32 KB, filled by TDM
  __shared__ float redA[NW];
  __shared__ float redB[NW];

  const int row  = blockIdx.x;
  const int tid  = threadIdx.x;
  const int wid  = tid >> 5;
  const int lane = tid & 31;
  const size_t base = (size_t)row * NCOLS;

  // ---- Tensor Data Mover: DMA this row's Lambda (32 KB) into LDS. --------
  // Descriptor per CDNA5 ISA 8.3/8.4: count=1, data_size=4B,
  // tile 8192x1, tensor_dim0=8192, stride=8192, type=2.  Uniform values ->
  // SGPRs; issued by wave 0 only (scalar branch skips other waves).
  {
    uint64_t ga = (uint64_t)(uintptr_t)(lam + base);
    u32 lds_base = (u32)(uintptr_t)(&lam_lds[0]);   // low 32 bits = LDS offset
    u32x4 g0;
    g0[0] = 1u;                                       // count=1, user mode
    g0[1] = lds_base;                                 // lds_addr
    g0[2] = (u32)(ga & 0xffffffffu);                  // global_addr[31:0]
    g0[3] = (u32)((ga >> 32) & 0x01ffffffu) | (2u << 30); // addr[56:32]|type=2
    u32x8 g1;
    g1[0] = 2u << 16;                 // wg_mask=0, data_size=2 (4 bytes)
    g1[1] = (u32)NCOLS << 16;         // tensor_dim0[15:0] in [31:16]
    g1[2] = 1u << 16;                 // tensor_dim0 hi=0 | tensor_dim1=1
    g1[3] = (u32)NCOLS << 16;         // tensor_dim1 hi=0 | tile_dim0=8192
    g1[4] = 1u;                       // tile_dim1=1, tile_dim2=0
    g1[5] = (u32)NCOLS;               // tensor_dim0_stride low32
    g1[6] = (u32)NCOLS << 16;         // stride hi=0 | tensor_dim1_stride lo
    g1[7] = 0u;
    if (tid < 32) {
      asm volatile("tensor_load_to_lds %0, %1" :: "s"(g0), "s"(g1) : "memory");
    }
  }

  // ---- residuals into registers (coalesced float4, lane-interleaved) -----
  const float4* yp4 = (const float4*)(yp + base);
  const float4* yt4 = (const float4*)(yt + base);
  float4 r[CH];
  float rmax = 0.0f;
#pragma unroll
  for (int i = 0; i < CH; ++i) {
    int idx4 = i * TPB + tid;
    float4 p = yp4[idx4];
    float4 t = yt4[idx4];
    float4 rv;
    rv.x = fabsf(t.x - p.x);
    rv.y = fabsf(t.y - p.y);
    rv.z = fabsf(t.z - p.z);
    rv.w = fabsf(t.w - p.w);
    r[i] = rv;
    rmax = fmaxf(rmax, fmaxf(fmaxf(rv.x, rv.y), fmaxf(rv.z, rv.w)));
  }

  // ---- block max -> initial eps ------------------------------------------
  rmax = wave_max(rmax);
  if (lane == 0) redA[wid] = rmax;
  __syncthreads();
  rmax = redA[0];
#pragma unroll
  for (int w = 1; w < NW; ++w) rmax = fmaxf(rmax, redA[w]);
  rmax = fmaxf(rmax, 1e-8f);

  float eps = rmax / (9.70406055f + 1e-8f);   // log(2N), N=8192
  const float invN = 1.0f / (float)NCOLS;

  // ---- Newton: mean(sinh(r/eps)) == 1 ------------------------------------
#pragma unroll 1
  for (int it = 0; it < 20; ++it) {
    float eps_safe = fmaxf(eps, 1e-8f);
    float inv_eps  = 1.0f / eps_safe;
    float s1 = 0.0f, s2 = 0.0f;
#pragma unroll
    for (int i = 0; i < CH; ++i) {
      sinhcosh_acc(r[i].x, inv_eps, s1, s2);
      sinhcosh_acc(r[i].y, inv_eps, s1, s2);
      sinhcosh_acc(r[i].z, inv_eps, s1, s2);
      sinhcosh_acc(r[i].w, inv_eps, s1, s2);
    }
    s1 = wave_sum(s1);
    s2 = wave_sum(s2);
    __syncthreads();                    // previous reads of redA/redB done
    if (lane == 0) { redA[wid] = s1; redB[wid] = s2; }
    __syncthreads();
    float t1 = 0.0f, t2 = 0.0f;
#pragma unroll
    for (int w = 0; w < NW; ++w) { t1 += redA[w]; t2 += redB[w]; }
    float val  = 0.5f * t1 * invN - 1.0f;
    float grad = -(0.5f * t2 * invN) * inv_eps;
    eps = eps - val / (grad - 1e-8f);   // uniform across block
  }

  // ---- beta, q, row max of q ---------------------------------------------
  float epsf = fmaxf(eps, 1e-8f);
  float beta = 1.0f / (epsf + 1e-6f);
  float qmax = 0.0f;
#pragma unroll
  for (int i = 0; i < CH; ++i) {
    qmax = fmaxf(qmax, sinh_q(r[i].x, beta));
    qmax = fmaxf(qmax, sinh_q(r[i].y, beta));
    qmax = fmaxf(qmax, sinh_q(r[i].z, beta));
    qmax = fmaxf(qmax, sinh_q(r[i].w, beta));
  }
  qmax = wave_max(qmax);

  // Lambda DMA must be complete before lam_lds reads after the barrier.
  asm volatile("s_wait_tensorcnt 0x0" ::: "memory");
  __syncthreads();
  if (lane == 0) redA[wid] = qmax;
  __syncthreads();
  qmax = redA[0];
#pragma unroll
  for (int w = 1; w < NW; ++w) qmax = fmaxf(qmax, redA[w]);
  float inv_qm = 1.0f / (qmax + 1e-20f);

  // ---- Lambda update + loss partial --------------------------------------
  float acc = 0.0f;
  const float4* lamv = (const float4*)lam_lds;
#pragma unroll
  for (int i = 0; i < CH; ++i) {
    int idx4 = i * TPB + tid;
    float4 L = lamv[idx4];
    float rr, q, li, up, tl;

    rr = r[i].x; q = sinh_q(rr, beta); li = 0.9f * (q * inv_qm) + 0.1f;
    up = 0.99f * L.x + 0.1f * li; out_lam[base + 4 * idx4 + 0] = up;
    tl = up * rr; acc += tl * tl;

    rr = r[i].y; q = sinh_q(rr, beta); li = 0.9f * (q * inv_qm) + 0.1f;
    up = 0.99f * L.y + 0.1f * li; out_lam[base + 4 * idx4 + 1] = up;
    tl = up * rr; acc += tl * tl;

    rr = r[i].z; q = sinh_q(rr, beta); li = 0.9f * (q * inv_qm) + 0.1f;
    up = 0.99f * L.z + 0.1f * li; out_lam[base + 4 * idx4 + 2] = up;
    tl = up * rr; acc += tl * tl;

    rr = r[i].w; q = sinh_q(rr, beta); li = 0.9f * (q * inv_qm) + 0.1f;
    up = 0.99f * L.w + 0.1f * li; out_lam[base + 4 * idx4 + 3] = up;
    tl = up * rr; acc += tl * tl;
  }

  acc = wave_sum(acc);
  __syncthreads();
  if (lane == 0) redA[wid] = acc;
  __syncthreads();
  if (tid == 0) {
    float s = 0.0f;
#pragma unroll
    for (int w = 0; w < NW; ++w) s += redA[w];
    row_loss[row] = s;
  }
}

// Deterministic fixed-order final reduction of 2048 row partials.
__global__ __launch_bounds__(TPB)
void loss_finalize(const float* __restrict__ row_loss, float* __restrict__ out)
{
  __shared__ float red[NW];
  const int tid = threadIdx.x;
  float a = 0.0f;
#pragma unroll
  for (int j = 0; j < BROWS / TPB; ++j) a += row_loss[tid + j * TPB];
  a = wave_sum(a);
  if ((tid & 31) == 0) red[tid >> 5] = a;
  __syncthreads();
  if (tid == 0) {
    float s = 0.0f;
#pragma unroll
    for (int w = 0; w < NW; ++w) s += red[w];
    out[0] = s * 0x1p-24f;   // / (2048*8192)
  }
}

extern "C" void kernel_launch(void* const* d_in, const int* in_sizes, int n_in,
                              void* d_out, int out_size, void* d_ws, size_t ws_size,
                              hipStream_t stream)
{
  const float* yp  = (const float*)d_in[0];
  const float* yt  = (const float*)d_in[1];
  const float* lam = (const float*)d_in[2];
  // d_in[3] = it (unused by cosh branch)
  float* out = (float*)d_out;          // [0]=loss, [1..] = updated_Lambda
  float* ws  = (float*)d_ws;           // BROWS row partial losses

  cosh_weight_rows<<<BROWS, TPB, 0, stream>>>(yp, yt, lam, out + 1, ws);
  loss_finalize<<<1, TPB, 0, stream>>>(ws, out);
}